// SGConvNet_51196010169027
// MI455X (gfx1250) — compile-verified
//
#include <hip/hip_runtime.h>

// SGConv (K=2) + Linear + sigmoid for MI455X (gfx1250, wave32).
// Pipeline:
//   deg = 1 + indegree           (atomic f32 add)
//   dinv = rsqrt(deg)
//   y1 = D^-1/2 (A+I) D^-1/2 x   (self-loop pass + edge scatter pass)
//   y2 = D^-1/2 (A+I) D^-1/2 y1
//   out = sigmoid(y2 @ W^T + b)  (V_WMMA_F32_16X16X4_F32)

#define N_NODES 100000
#define N_EDGES 800000
#define IN_C 96
#define OUT_C 64

typedef __attribute__((ext_vector_type(2))) float v2f;
typedef __attribute__((ext_vector_type(4))) float v4f;
typedef __attribute__((ext_vector_type(8))) float v8f;

// ---------------------------------------------------------------- degree ----
__global__ __launch_bounds__(256) void k_init_deg(float* __restrict__ deg) {
  int i = blockIdx.x * 256 + threadIdx.x;
  if (i < N_NODES) deg[i] = 1.0f;  // self-loop contribution
}

__global__ __launch_bounds__(256) void k_accum_deg(const long long* __restrict__ col,
                                                   float* __restrict__ deg) {
  int e = blockIdx.x * 256 + threadIdx.x;
  if (e < N_EDGES) atomicAdd(&deg[(int)col[e]], 1.0f);
}

__global__ __launch_bounds__(256) void k_dinv(const float* __restrict__ deg,
                                              float* __restrict__ dinv) {
  int i = blockIdx.x * 256 + threadIdx.x;
  if (i < N_NODES) dinv[i] = rsqrtf(deg[i]);  // deg >= 1 always
}

// ------------------------------------------------------------ propagation ----
// y[i][:] = dinv[i]^2 * x[i][:]   (self-loop term), float4 vectorized.
__global__ __launch_bounds__(256) void k_self_init(const float* __restrict__ xin,
                                                   const float* __restrict__ dinv,
                                                   float* __restrict__ y) {
  int i = blockIdx.x * 256 + threadIdx.x;          // float4 index
  const int total = N_NODES * (IN_C / 4);          // 2,400,000 (exact multiple of 256: yes, 9375 blocks)
  if (i >= total) return;
  int node = i / (IN_C / 4);                       // 24 float4 per row
  float s = dinv[node];
  s = s * s;
  v4f v = ((const v4f*)xin)[i];
  ((v4f*)y)[i] = v * s;
}

// One wave32 per edge: lane handles channels {lane, lane+32, lane+64}.
// y[col][c] += dinv[row]*dinv[col] * x[row][c]
__global__ __launch_bounds__(256) void k_scatter(const long long* __restrict__ row,
                                                 const long long* __restrict__ col,
                                                 const float* __restrict__ dinv,
                                                 const float* __restrict__ xin,
                                                 float* __restrict__ y) {
  int e = blockIdx.x * 8 + (threadIdx.x >> 5);     // 100000 blocks * 8 waves == N_EDGES
  int lane = threadIdx.x & 31;
  int r = (int)row[e];
  int c = (int)col[e];
  float nrm = dinv[r] * dinv[c];
  const float* xs = xin + (size_t)r * IN_C;
  float* yd = y + (size_t)c * IN_C;
  atomicAdd(&yd[lane],      nrm * xs[lane]);
  atomicAdd(&yd[lane + 32], nrm * xs[lane + 32]);
  atomicAdd(&yd[lane + 64], nrm * xs[lane + 64]);
}

// --------------------------------------------------- WMMA GEMM + sigmoid ----
// out[N x 64] = sigmoid(X[N x 96] @ W^T + b), W stored [64 x 96] row-major.
// One wave per 16-row tile; computes all four 16-col tiles (A reused 4x).
// V_WMMA_F32_16X16X4_F32 operand layouts (wave32):
//   A 16x4 : lane holds M = lane&15, K = 2*(lane>>4) + {0,1}  -> contiguous float2
//   B 4x16 : lane holds N = lane&15, K = 2*(lane>>4) + {0,1}  -> contiguous float2 of W[n][k]
//   D 16x16: VGPR r holds M = r + 8*(lane>>4), N = lane&15
__global__ __launch_bounds__(256) void k_gemm_sigmoid(const float* __restrict__ X,
                                                      const float* __restrict__ W,
                                                      const float* __restrict__ bias,
                                                      float* __restrict__ out) {
  int wave = (blockIdx.x * 256 + threadIdx.x) >> 5;
  if (wave >= N_NODES / 16) return;                // wave-uniform guard: EXEC all-1s inside
  int lane = threadIdx.x & 31;
  int half = lane >> 4;
  int idx  = lane & 15;
  int m0 = wave * 16;

  const float* arow = X + (size_t)(m0 + idx) * IN_C + half * 2;
  const float* brow = W + (size_t)idx * IN_C + half * 2;  // n-tile 0; tiles j offset by 16*j*IN_C

  v8f acc0 = {}, acc1 = {}, acc2 = {}, acc3 = {};
#pragma unroll
  for (int k0 = 0; k0 < IN_C; k0 += 4) {
    v2f a   = *(const v2f*)(arow + k0);
    v2f bb0 = *(const v2f*)(brow + k0);
    v2f bb1 = *(const v2f*)(brow + 16 * IN_C + k0);
    v2f bb2 = *(const v2f*)(brow + 32 * IN_C + k0);
    v2f bb3 = *(const v2f*)(brow + 48 * IN_C + k0);
    acc0 = __builtin_amdgcn_wmma_f32_16x16x4_f32(false, a, false, bb0, (short)0, acc0, false, false);
    acc1 = __builtin_amdgcn_wmma_f32_16x16x4_f32(false, a, false, bb1, (short)0, acc1, false, false);
    acc2 = __builtin_amdgcn_wmma_f32_16x16x4_f32(false, a, false, bb2, (short)0, acc2, false, false);
    acc3 = __builtin_amdgcn_wmma_f32_16x16x4_f32(false, a, false, bb3, (short)0, acc3, false, false);
  }

  float bv0 = bias[idx];
  float bv1 = bias[16 + idx];
  float bv2 = bias[32 + idx];
  float bv3 = bias[48 + idx];

#pragma unroll
  for (int r = 0; r < 8; ++r) {
    int m = m0 + half * 8 + r;
    float* o = out + (size_t)m * OUT_C + idx;
    o[0]  = 1.0f / (1.0f + expf(-(acc0[r] + bv0)));
    o[16] = 1.0f / (1.0f + expf(-(acc1[r] + bv1)));
    o[32] = 1.0f / (1.0f + expf(-(acc2[r] + bv2)));
    o[48] = 1.0f / (1.0f + expf(-(acc3[r] + bv3)));
  }
}

// ------------------------------------------------------------------ launch ----
extern "C" void kernel_launch(void* const* d_in, const int* in_sizes, int n_in,
                              void* d_out, int out_size, void* d_ws, size_t ws_size,
                              hipStream_t stream) {
  (void)in_sizes; (void)n_in; (void)out_size; (void)ws_size;

  const float*     x  = (const float*)d_in[0];           // [N, 96]
  const long long* ei = (const long long*)d_in[1];       // [2, E] int64
  const float*     W  = (const float*)d_in[2];           // [64, 96]
  const float*     b  = (const float*)d_in[3];           // [64]
  float*           out = (float*)d_out;                  // [N, 64]

  const long long* row = ei;            // sources
  const long long* col = ei + N_EDGES;  // destinations

  // Workspace layout (floats): deg[N] | dinv[N] | y1[N*96] | y2[N*96]
  float* deg  = (float*)d_ws;
  float* dinv = deg + N_NODES;
  float* y1   = dinv + N_NODES;
  float* y2   = y1 + (size_t)N_NODES * IN_C;

  const int nodeBlocks  = (N_NODES + 255) / 256;                 // 391
  const int edgeBlocks  = (N_EDGES + 255) / 256;                 // 3125
  const int featBlocks  = (N_NODES * (IN_C / 4) + 255) / 256;    // 9375
  const int scatBlocks  = N_EDGES / 8;                           // 100000 (exact)
  const int gemmBlocks  = (N_NODES / 16 + 7) / 8;                // 782

  // Degree + normalization
  k_init_deg<<<nodeBlocks, 256, 0, stream>>>(deg);
  k_accum_deg<<<edgeBlocks, 256, 0, stream>>>(col, deg);
  k_dinv<<<nodeBlocks, 256, 0, stream>>>(deg, dinv);

  // Hop 1: y1 = norm-prop(x)
  k_self_init<<<featBlocks, 256, 0, stream>>>(x, dinv, y1);
  k_scatter<<<scatBlocks, 256, 0, stream>>>(row, col, dinv, x, y1);

  // Hop 2: y2 = norm-prop(y1)
  k_self_init<<<featBlocks, 256, 0, stream>>>(y1, dinv, y2);
  k_scatter<<<scatBlocks, 256, 0, stream>>>(row, col, dinv, y1, y2);

  // Linear + sigmoid via fp32 WMMA
  k_gemm_sigmoid<<<gemmBlocks, 256, 0, stream>>>(y2, W, b, out);
}